// Net_26156350833001
// MI455X (gfx1250) — compile-verified
//
#include <hip/hip_runtime.h>
#include <hip/hip_bf16.h>
#include <math.h>

typedef __attribute__((ext_vector_type(2))) float v2f;
typedef __attribute__((ext_vector_type(8))) float v8f;

// ---------------------------------------------------------------------------
// Degree accumulation: deg[c] += 1 per edge (self-loop +1 folded into dinv).
// ---------------------------------------------------------------------------
__global__ void deg_kernel(const int* __restrict__ col, float* __restrict__ deg, int E) {
    int t = blockIdx.x * blockDim.x + threadIdx.x;
    if (t < E) atomicAdd(&deg[col[t]], 1.0f);
}

__global__ void dinv_kernel(float* __restrict__ deg, int n) {
    int i = blockIdx.x * blockDim.x + threadIdx.x;
    if (i < n) {
        float d = deg[i] + 1.0f;            // + self loop
        deg[i] = __frsqrt_rn(d);            // deg >= 1 always
    }
}

// ---------------------------------------------------------------------------
// GEMM1: h[n x 16] = x[n x 512] @ W1[512 x 16] using V_WMMA_F32_16X16X4_F32.
// One wave -> one 16x16 output tile; 128 K-steps of 4.
// A frag: lane<16 holds K=k0,k0+1 ; lane>=16 holds K=k0+2,k0+3 (ISA 7.12.2).
// B frag from LDS-staged W1 (32KB).
// ---------------------------------------------------------------------------
__global__ __launch_bounds__(256) void gemm1_wmma(const float* __restrict__ x,
                                                  const float* __restrict__ W1,
                                                  float* __restrict__ h, int n) {
    __shared__ float w[512 * 16];
    for (int i = threadIdx.x; i < 512 * 16; i += 256) w[i] = W1[i];
    __syncthreads();

    const int wave = threadIdx.x >> 5;
    const int lane = threadIdx.x & 31;
    const int m0   = (blockIdx.x * 8 + wave) * 16;
    if (m0 >= n) return;                    // wave-uniform: EXEC stays all-ones

    const int nlo  = lane & 15;
    const int klo  = (lane >> 4) * 2;       // 0 for lanes 0-15, 2 for 16-31
    int mrow = m0 + nlo;
    if (mrow >= n) mrow = n - 1;            // clamp loads for ragged tail
    const float* xrow = x + (size_t)mrow * 512 + klo;

    v8f acc = {};
    for (int kb = 0; kb < 512; kb += 64) {
        __builtin_prefetch(xrow + kb + 64, 0, 0);   // global_prefetch_b8
        #pragma unroll
        for (int k0 = kb; k0 < kb + 64; k0 += 4) {
            v2f a = *(const v2f*)(xrow + k0);       // x[m][k0+klo], x[m][k0+klo+1]
            v2f b;
            b.x = w[(k0 + klo) * 16 + nlo];         // W1[k0+klo    ][nlo]
            b.y = w[(k0 + klo + 1) * 16 + nlo];     // W1[k0+klo + 1][nlo]
            acc = __builtin_amdgcn_wmma_f32_16x16x4_f32(
                false, a, false, b, (short)0, acc, false, false);
        }
    }

    const int rbase = m0 + (lane >> 4) * 8;
    #pragma unroll
    for (int r = 0; r < 8; ++r) {
        int mr = rbase + r;
        if (mr < n) h[(size_t)mr * 16 + nlo] = acc[r];
    }
}

// ---------------------------------------------------------------------------
// Layer-1 edge aggregation: 16 lanes per edge (one per feature).
// out1[c][f] += h[r][f] * dinv[r]*dinv[c]
// ---------------------------------------------------------------------------
__global__ void agg1_kernel(const int* __restrict__ row, const int* __restrict__ col,
                            const float* __restrict__ h, const float* __restrict__ dinv,
                            float* __restrict__ out1, int E) {
    long long t = (long long)blockIdx.x * blockDim.x + threadIdx.x;
    long long e = t >> 4;
    int f = (int)(t & 15);
    if (e >= E) return;
    int r = row[e], c = col[e];
    float norm = dinv[r] * dinv[c];
    atomicAdd(&out1[(size_t)c * 16 + f], h[(size_t)r * 16 + f] * norm);
}

// self-loop term + bias + ReLU (in place on out1 -> activated h)
__global__ void finish1_kernel(float* __restrict__ out1, const float* __restrict__ h,
                               const float* __restrict__ dinv, const float* __restrict__ b1,
                               int n) {
    int t = blockIdx.x * blockDim.x + threadIdx.x;
    int i = t >> 4, f = t & 15;
    if (i >= n) return;
    float d = dinv[i];
    float v = out1[(size_t)i * 16 + f] + h[(size_t)i * 16 + f] * d * d + b1[f];
    out1[(size_t)i * 16 + f] = fmaxf(v, 0.0f);
}

// ---------------------------------------------------------------------------
// GEMM2: h2[n x 8] = hact[n x 16] @ W2pad[16 x 16] (cols 7..15 zero), 4 K-steps.
// ---------------------------------------------------------------------------
__global__ __launch_bounds__(256) void gemm2_wmma(const float* __restrict__ hact,
                                                  const float* __restrict__ W2,
                                                  float* __restrict__ h2, int n) {
    __shared__ float w[16 * 16];
    {
        int f = threadIdx.x >> 4, j = threadIdx.x & 15;
        if (threadIdx.x < 256) w[f * 16 + j] = (j < 7) ? W2[f * 7 + j] : 0.0f;
    }
    __syncthreads();

    const int wave = threadIdx.x >> 5;
    const int lane = threadIdx.x & 31;
    const int m0   = (blockIdx.x * 8 + wave) * 16;
    if (m0 >= n) return;

    const int nlo = lane & 15;
    const int klo = (lane >> 4) * 2;
    int mrow = m0 + nlo;
    if (mrow >= n) mrow = n - 1;
    const float* arow = hact + (size_t)mrow * 16 + klo;

    v8f acc = {};
    #pragma unroll
    for (int k0 = 0; k0 < 16; k0 += 4) {
        v2f a = *(const v2f*)(arow + k0);
        v2f b;
        b.x = w[(k0 + klo) * 16 + nlo];
        b.y = w[(k0 + klo + 1) * 16 + nlo];
        acc = __builtin_amdgcn_wmma_f32_16x16x4_f32(
            false, a, false, b, (short)0, acc, false, false);
    }

    const int rbase = m0 + (lane >> 4) * 8;
    #pragma unroll
    for (int r = 0; r < 8; ++r) {
        int mr = rbase + r;
        if (mr < n && nlo < 8) h2[(size_t)mr * 8 + nlo] = acc[r];  // col 7 is 0 pad
    }
}

// Layer-2 edge aggregation: 8 lanes per edge (7 real features + zero pad lane).
__global__ void agg2_kernel(const int* __restrict__ row, const int* __restrict__ col,
                            const float* __restrict__ h2, const float* __restrict__ dinv,
                            float* __restrict__ out2, int E) {
    long long t = (long long)blockIdx.x * blockDim.x + threadIdx.x;
    long long e = t >> 3;
    int f = (int)(t & 7);
    if (e >= E) return;
    int r = row[e], c = col[e];
    float norm = dinv[r] * dinv[c];
    atomicAdd(&out2[(size_t)c * 8 + f], h2[(size_t)r * 8 + f] * norm);
}

// self-loop + bias, then logits + log_softmax written to d_out.
__global__ void finish2_kernel(const float* __restrict__ out2, const float* __restrict__ h2,
                               const float* __restrict__ dinv, const float* __restrict__ b2,
                               float* __restrict__ out, int n) {
    int i = blockIdx.x * blockDim.x + threadIdx.x;
    if (i >= n) return;
    float d  = dinv[i];
    float dd = d * d;
    float l[7];
    float m = -INFINITY;
    #pragma unroll
    for (int j = 0; j < 7; ++j) {
        l[j] = out2[(size_t)i * 8 + j] + h2[(size_t)i * 8 + j] * dd + b2[j];
        m = fmaxf(m, l[j]);
    }
    float s = 0.0f;
    #pragma unroll
    for (int j = 0; j < 7; ++j) s += __expf(l[j] - m);
    float lse = m + __logf(s);
    #pragma unroll
    for (int j = 0; j < 7; ++j) {
        out[(size_t)i * 7 + j]                 = l[j];          // logits
        out[(size_t)n * 7 + (size_t)i * 7 + j] = l[j] - lse;    // log_softmax
    }
}

// ---------------------------------------------------------------------------
extern "C" void kernel_launch(void* const* d_in, const int* in_sizes, int n_in,
                              void* d_out, int out_size, void* d_ws, size_t ws_size,
                              hipStream_t stream) {
    const float* x   = (const float*)d_in[0];
    const int*   ei  = (const int*)d_in[1];     // [2, E]
    const float* W1  = (const float*)d_in[2];
    const float* b1  = (const float*)d_in[3];
    const float* W2  = (const float*)d_in[4];
    const float* b2  = (const float*)d_in[5];
    float*       out = (float*)d_out;

    const int n = in_sizes[0] / 512;
    const int E = in_sizes[1] / 2;
    const int* row = ei;
    const int* col = ei + E;

    // workspace layout (rounded node count keeps 16B alignment everywhere)
    size_t nn = ((size_t)n + 255) & ~(size_t)255;
    float* deg  = (float*)d_ws;          // n         -> becomes dinv in place
    float* h1   = deg  + nn;             // n*16
    float* out1 = h1   + nn * 16;        // n*16      -> becomes activated h
    float* h2   = out1 + nn * 16;        // n*8
    float* out2 = h2   + nn * 8;         // n*8
    (void)ws_size; (void)n_in; (void)out_size;

    hipMemsetAsync(deg,  0, (size_t)n * sizeof(float), stream);
    hipMemsetAsync(out1, 0, (size_t)n * 16 * sizeof(float), stream);
    hipMemsetAsync(out2, 0, (size_t)n * 8  * sizeof(float), stream);

    // degrees + normalization
    deg_kernel<<<(E + 255) / 256, 256, 0, stream>>>(col, deg, E);
    dinv_kernel<<<(n + 255) / 256, 256, 0, stream>>>(deg, n);

    // layer 1
    int tiles  = (n + 15) / 16;
    int gblocks = (tiles + 7) / 8;
    gemm1_wmma<<<gblocks, 256, 0, stream>>>(x, W1, h1, n);

    long long a1 = (long long)E * 16;
    agg1_kernel<<<(unsigned)((a1 + 255) / 256), 256, 0, stream>>>(row, col, h1, deg, out1, E);

    long long f1 = (long long)n * 16;
    finish1_kernel<<<(unsigned)((f1 + 255) / 256), 256, 0, stream>>>(out1, h1, deg, b1, n);

    // layer 2
    gemm2_wmma<<<gblocks, 256, 0, stream>>>(out1, W2, h2, n);

    long long a2 = (long long)E * 8;
    agg2_kernel<<<(unsigned)((a2 + 255) / 256), 256, 0, stream>>>(row, col, h2, deg, out2, E);

    finish2_kernel<<<(n + 255) / 256, 256, 0, stream>>>(out2, h2, deg, b2, out, n);
}